// GraphAttentionLayer_28724741276291
// MI455X (gfx1250) — compile-verified
//
#include <hip/hip_runtime.h>
#include <math.h>

// GAT forward for MI455X (gfx1250, wave32).
// Key algebraic fusion: (edge @ edge_map_w^T + b) @ edge_lin_w^T
//   == edge @ (edge_map_w^T @ edge_lin_w^T) + (b . edge_lin_w)
// so the 15.7 GFLOP edge MLP collapses to a 100-length dot per edge and the
// whole problem is bound by streaming edge_embedding (105 MB ~ 4.5us @ 23.3TB/s).
// GEMMs (x@W, h@V^T, att@h) run on V_WMMA_F32_16X16X4_F32 (fp32-exact matrix pipe).
// Each wave computes a 32x16 macro-tile: 2 independent WMMAs per k-step share
// one B fetch (halves strided B traffic, no EXEC divergence in inner loops —
// partial N tile handled by clamped addresses since WMMA columns don't mix).

constexpr int Bb  = 16;
constexpr int Ll  = 128;
constexpr int DIN = 768;
constexpr int DHh = 300;
constexpr int DEe = 100;
constexpr float ALPHA = 0.2f;
constexpr float NEGV  = -9.0e15f;

typedef float v2f __attribute__((ext_vector_type(2)));
typedef float v8f __attribute__((ext_vector_type(8)));

__device__ __forceinline__ v8f wmma_f32_16x16x4(v2f a, v2f b, v8f c) {
  // 8 args: (neg_a, A, neg_b, B, c_mod, C, reuse_a, reuse_b)
  return __builtin_amdgcn_wmma_f32_16x16x4_f32(false, a, false, b, (short)0, c,
                                               false, false);
}

// ---------------------------------------------------------------------------
// prep: V scatter (128x300), ve = edge_map_w^T @ edge_lin_w (100), be scalar
// ---------------------------------------------------------------------------
__global__ __launch_bounds__(256)
void prep_kernel(const float* __restrict__ a,    // (2*DH, 1) flat
                 const float* __restrict__ emw,  // (DH, DE)
                 const float* __restrict__ emb,  // (DH)
                 const float* __restrict__ elw,  // (1, DH)
                 const float* __restrict__ elb,  // (1)
                 float* __restrict__ Vm,         // (L, DH)
                 float* __restrict__ ve,         // (DE)
                 float* __restrict__ be) {       // (1)
  const int t = threadIdx.x;
  for (int i = t; i < Ll * DHh; i += 256) Vm[i] = 0.0f;
  for (int d = t; d < DEe; d += 256) {
    float s = 0.0f;
    for (int h = 0; h < DHh; ++h) s += emw[h * DEe + d] * elw[h];
    ve[d] = s;
  }
  if (t == 0) {
    float s = 0.0f;
    for (int h = 0; h < DHh; ++h) s += emb[h] * elw[h];
    *be = s + elb[0];
  }
  __syncthreads();
  // V[i, (i*DH + j)//L] += a_first[j]
  for (int i = t; i < Ll * DHh; i += 256) {
    const int row = i / DHh;
    const int j   = i - row * DHh;
    const int c   = (row * DHh + j) >> 7;  // / 128
    atomicAdd(&Vm[row * DHh + c], a[j]);
  }
}

// ---------------------------------------------------------------------------
// h = x @ W : (2048 x 768) @ (768 x 300). One 32x16 macro-tile per wave.
// ---------------------------------------------------------------------------
__global__ __launch_bounds__(32)
void gemm_xw(const float* __restrict__ X, const float* __restrict__ W,
             float* __restrict__ H) {
  const int nt  = blockIdx.x;         // 0..18 (N tiles, last partial)
  const int mt0 = blockIdx.y * 2;     // 0..126 step 2
  const int lane = threadIdx.x;
  const int half = lane >> 4;
  const int l16  = lane & 15;
  const int n  = nt * 16 + l16;                // B/D column owned by this lane
  const int nc = (n < DHh) ? n : (DHh - 1);    // clamp: invalid cols never stored
  const float* xr0 = X + (size_t)(mt0 * 16 + l16) * DIN;
  const float* xr1 = xr0 + (size_t)16 * DIN;
  v8f acc0 = {}, acc1 = {};
  for (int k0 = 0; k0 < DIN; k0 += 4) {
    const int ka = k0 + 2 * half;     // ISA layout: K = vgpr + 2*(lane>=16)
    v2f a0; a0.x = xr0[ka]; a0.y = xr0[ka + 1];
    v2f a1; a1.x = xr1[ka]; a1.y = xr1[ka + 1];
    v2f bv; bv.x = W[ka * DHh + nc]; bv.y = W[(ka + 1) * DHh + nc];
    acc0 = wmma_f32_16x16x4(a0, bv, acc0);
    acc1 = wmma_f32_16x16x4(a1, bv, acc1);
  }
  if (n < DHh) {
#pragma unroll
    for (int v = 0; v < 8; ++v) {
      const int mm = mt0 * 16 + v + 8 * half;
      H[(size_t)mm * DHh + n]        = acc0[v];
      H[(size_t)(mm + 16) * DHh + n] = acc1[v];
    }
  }
}

// ---------------------------------------------------------------------------
// e2[b,j] = dot(h[b,j,:], a[DH:2DH]) — one wave per row
// ---------------------------------------------------------------------------
__global__ __launch_bounds__(256)
void e2_kernel(const float* __restrict__ H, const float* __restrict__ a,
               float* __restrict__ E2) {
  const int wave = (blockIdx.x * blockDim.x + threadIdx.x) >> 5;  // 0..2047
  const int lane = threadIdx.x & 31;
  const float* hr = H + (size_t)wave * DHh;
  float s = 0.0f;
  for (int d = lane; d < DHh; d += 32) s += hr[d] * a[DHh + d];
#pragma unroll
  for (int off = 16; off; off >>= 1) s += __shfl_xor(s, off, 32);
  if (lane == 0) E2[wave] = s;
}

// ---------------------------------------------------------------------------
// e1[b] = h[b] @ V^T : (128 x 300) @ (300 x 128), K=300 = 75*4
// 32x16 macro-tile per wave; B[k,n] = V[n,k].
// ---------------------------------------------------------------------------
__global__ __launch_bounds__(32)
void gemm_e1(const float* __restrict__ H, const float* __restrict__ Vm,
             float* __restrict__ E1) {
  const int nt  = blockIdx.x;         // 0..7 (j tiles)
  const int mt0 = blockIdx.y * 2;     // 0..6 step 2 (i tiles)
  const int b   = blockIdx.z;         // 0..15
  const int lane = threadIdx.x;
  const int half = lane >> 4;
  const int l16  = lane & 15;
  const float* Hb = H + (size_t)b * Ll * DHh;
  const int n = nt * 16 + l16;
  const float* hr0 = Hb + (size_t)(mt0 * 16 + l16) * DHh;
  const float* hr1 = hr0 + (size_t)16 * DHh;
  const float* vr  = Vm + (size_t)n * DHh;
  v8f acc0 = {}, acc1 = {};
  for (int k0 = 0; k0 < DHh; k0 += 4) {
    const int ka = k0 + 2 * half;
    v2f a0; a0.x = hr0[ka]; a0.y = hr0[ka + 1];
    v2f a1; a1.x = hr1[ka]; a1.y = hr1[ka + 1];
    v2f bv; bv.x = vr[ka];  bv.y = vr[ka + 1];
    acc0 = wmma_f32_16x16x4(a0, bv, acc0);
    acc1 = wmma_f32_16x16x4(a1, bv, acc1);
  }
  float* ob = E1 + (size_t)b * Ll * Ll;
#pragma unroll
  for (int v = 0; v < 8; ++v) {
    const int mm = mt0 * 16 + v + 8 * half;
    ob[mm * Ll + n]        = acc0[v];
    ob[(mm + 16) * Ll + n] = acc1[v];
  }
}

// ---------------------------------------------------------------------------
// scores: stream edge_embedding (the 105MB term), fuse sigmoid-gate, leaky
// relu, adj mask, row softmax. One block (8 waves) per (b,i) row of 128 j's.
// Each wave: coalesced 400B float4 row load, shuffle-reduced dot with ve.
// ---------------------------------------------------------------------------
__global__ __launch_bounds__(256)
void scores_kernel(const float* __restrict__ EDGE, const float* __restrict__ E1,
                   const float* __restrict__ E2, const int* __restrict__ ADJ,
                   const float* __restrict__ VE, const float* __restrict__ BE,
                   float* __restrict__ ATT) {
  const int i = blockIdx.x;
  const int b = blockIdx.y;
  const int t = threadIdx.x;
  const int wave = t >> 5;
  const int lane = t & 31;
  __shared__ float lg[Ll];
  __shared__ float s_red[2];

  float4 vv = make_float4(0.f, 0.f, 0.f, 0.f);
  if (lane < 25) vv = ((const float4*)VE)[lane];   // DE=100 = 25 float4
  const float be = *BE;
  const int rowbase = (b * Ll + i) * Ll;

  for (int jj = 0; jj < 16; ++jj) {
    const int j = wave * 16 + jj;
    const float* erow = EDGE + (size_t)(rowbase + j) * DEe;
    if (jj + 1 < 16)
      __builtin_prefetch(EDGE + (size_t)(rowbase + j + 1) * DEe, 0, 1);
    float4 ev = make_float4(0.f, 0.f, 0.f, 0.f);
    if (lane < 25) ev = ((const float4*)erow)[lane];
    float p = ev.x * vv.x + ev.y * vv.y + ev.z * vv.z + ev.w * vv.w;
#pragma unroll
    for (int off = 16; off; off >>= 1) p += __shfl_xor(p, off, 32);
    if (lane == 0) {
      const float ew = 1.0f / (1.0f + expf(-(p + be)));
      const float e  = E1[rowbase + j] + E2[b * Ll + j];
      const float lr = (e > 0.0f) ? e : ALPHA * e;
      lg[j] = (ADJ[rowbase + j] > 0) ? lr * ew : NEGV;
    }
  }
  __syncthreads();

  if (t < 32) {
    float m = -INFINITY;
    for (int j = lane; j < Ll; j += 32) m = fmaxf(m, lg[j]);
#pragma unroll
    for (int off = 16; off; off >>= 1) m = fmaxf(m, __shfl_xor(m, off, 32));
    float s = 0.0f;
    for (int j = lane; j < Ll; j += 32) s += expf(lg[j] - m);
#pragma unroll
    for (int off = 16; off; off >>= 1) s += __shfl_xor(s, off, 32);
    if (lane == 0) { s_red[0] = m; s_red[1] = s; }
  }
  __syncthreads();

  if (t < Ll) {
    const float inv = 1.0f / s_red[1];
    ATT[rowbase + t] = expf(lg[t] - s_red[0]) * inv;
  }
}

// ---------------------------------------------------------------------------
// out[b] = h[b] + att[b] @ h[b] : (128 x 128) @ (128 x 300)
// 32x16 macro-tile per wave; clamped B columns for the partial N tile.
// ---------------------------------------------------------------------------
__global__ __launch_bounds__(32)
void gemm_out(const float* __restrict__ ATT, const float* __restrict__ H,
              float* __restrict__ OUT) {
  const int nt  = blockIdx.x;         // 0..18
  const int mt0 = blockIdx.y * 2;     // 0..6 step 2
  const int b   = blockIdx.z;         // 0..15
  const int lane = threadIdx.x;
  const int half = lane >> 4;
  const int l16  = lane & 15;
  const float* Ab = ATT + (size_t)b * Ll * Ll;
  const float* Hb = H   + (size_t)b * Ll * DHh;
  const int n  = nt * 16 + l16;
  const int nc = (n < DHh) ? n : (DHh - 1);
  const float* ar0 = Ab + (size_t)(mt0 * 16 + l16) * Ll;
  const float* ar1 = ar0 + (size_t)16 * Ll;
  v8f acc0 = {}, acc1 = {};
  for (int k0 = 0; k0 < Ll; k0 += 4) {
    const int ka = k0 + 2 * half;
    v2f a0; a0.x = ar0[ka]; a0.y = ar0[ka + 1];
    v2f a1; a1.x = ar1[ka]; a1.y = ar1[ka + 1];
    v2f bv; bv.x = Hb[ka * DHh + nc]; bv.y = Hb[(ka + 1) * DHh + nc];
    acc0 = wmma_f32_16x16x4(a0, bv, acc0);
    acc1 = wmma_f32_16x16x4(a1, bv, acc1);
  }
  if (n < DHh) {
    float* Ob = OUT + (size_t)b * Ll * DHh;
#pragma unroll
    for (int v = 0; v < 8; ++v) {
      const int mm = mt0 * 16 + v + 8 * half;
      Ob[mm * DHh + n]        = Hb[mm * DHh + n] + acc0[v];
      Ob[(mm + 16) * DHh + n] = Hb[(mm + 16) * DHh + n] + acc1[v];
    }
  }
}

// ---------------------------------------------------------------------------
extern "C" void kernel_launch(void* const* d_in, const int* in_sizes, int n_in,
                              void* d_out, int out_size, void* d_ws, size_t ws_size,
                              hipStream_t stream) {
  (void)in_sizes; (void)n_in; (void)out_size; (void)ws_size;
  const float* x    = (const float*)d_in[0];  // (B,L,DIN)
  const int*   adj  = (const int*)  d_in[1];  // (B,L,L)
  const float* edge = (const float*)d_in[2];  // (B,L,L,DE)
  const float* W    = (const float*)d_in[3];  // (DIN,DH)
  const float* a    = (const float*)d_in[4];  // (2DH,1)
  const float* emw  = (const float*)d_in[5];  // (DH,DE)
  const float* emb  = (const float*)d_in[6];  // (DH)
  const float* elw  = (const float*)d_in[7];  // (1,DH)
  const float* elb  = (const float*)d_in[8];  // (1)
  float* out = (float*)d_out;

  // workspace layout (floats): total ~4.72 MB
  float* ws = (float*)d_ws;
  float* H  = ws;              // B*L*DH  = 614400
  float* Vm = H  + 614400;     // L*DH    = 38400
  float* E1 = Vm + 38400;      // B*L*L   = 262144
  float* E2 = E1 + 262144;     // B*L     = 2048
  float* VE = E2 + 2048;       // DE pad  = 128
  float* BE = VE + 128;        // 8
  float* AT = BE + 8;          // B*L*L   = 262144

  prep_kernel  <<<1, 256, 0, stream>>>(a, emw, emb, elw, elb, Vm, VE, BE);
  gemm_xw      <<<dim3(19, 64),    32, 0, stream>>>(x, W, H);
  e2_kernel    <<<dim3(256),      256, 0, stream>>>(H, a, E2);
  gemm_e1      <<<dim3(8, 4, 16),  32, 0, stream>>>(H, Vm, E1);
  scores_kernel<<<dim3(128, 16),  256, 0, stream>>>(edge, E1, E2, adj, VE, BE, AT);
  gemm_out     <<<dim3(19, 4, 16), 32, 0, stream>>>(AT, H, out);
}